// DiffusionGraphConv_78374563217429
// MI455X (gfx1250) — compile-verified
//
#include <hip/hip_runtime.h>
#include <hip/hip_bf16.h>

#define NODES 4096
#define FDIM  128          // INPUT_DIM + HID_DIM
#define BATCH 32
#define OUTD  64
#define ROWS_TOTAL (BATCH * NODES)          // 131072 GEMM rows
#define BUF_ELEMS  ((size_t)BATCH * NODES * FDIM)   // 16,777,216 floats = 64 MB

typedef float v2f __attribute__((ext_vector_type(2)));
typedef float v8f __attribute__((ext_vector_type(8)));

// ---------------- concat: x0[b][n][0:64]=inputs, [64:128]=state ----------------
__global__ __launch_bounds__(256) void k_concat(const float* __restrict__ in,
                                                const float* __restrict__ st,
                                                float* __restrict__ x0) {
  size_t g = (size_t)blockIdx.x * blockDim.x + threadIdx.x;   // float4 index
  size_t bn = g >> 5;                                          // (b*N+n)
  int q = (int)(g & 31);                                       // 32 float4 per row
  const float4* src = (q < 16) ? ((const float4*)in + bn * 16 + q)
                               : ((const float4*)st + bn * 16 + (q - 16));
  ((float4*)x0)[g] = *src;
}

// ---------------- zero a 64MB buffer ----------------
__global__ __launch_bounds__(256) void k_zero(float* __restrict__ p) {
  size_t g = (size_t)blockIdx.x * blockDim.x + threadIdx.x;
  ((float4*)p)[g] = make_float4(0.f, 0.f, 0.f, 0.f);
}

// ---------------- SpMM scatter: y[:,row,:] += val * x[:,col,:] ----------------
__global__ __launch_bounds__(256) void k_spmm(const int* __restrict__ rows,
                                              const int* __restrict__ cols,
                                              const float* __restrict__ vals,
                                              const float* __restrict__ x,
                                              float* __restrict__ y) {
  int e = blockIdx.x;
  int row = rows[e];
  int col = cols[e];
  float v = vals[e];
  int f  = threadIdx.x & 127;
  int b0 = threadIdx.x >> 7;                // 0 or 1
  const size_t bstride = (size_t)NODES * FDIM;
  const float* xs = x + (size_t)col * FDIM + f;
  float*       yd = y + (size_t)row * FDIM + f;
  for (int b = b0; b < BATCH; b += 2) {
    float c = v * xs[(size_t)b * bstride];
    // native f32 atomic add, no return -> global_atomic_add_f32 (L2-resident RMW)
    __hip_atomic_fetch_add(yd + (size_t)b * bstride, c,
                           __ATOMIC_RELAXED, __HIP_MEMORY_SCOPE_AGENT);
  }
}

// ---------------- Chebyshev combine: z = 2*z - x0 ----------------
__global__ __launch_bounds__(256) void k_cheb(float* __restrict__ z,
                                              const float* __restrict__ x0) {
  size_t g = (size_t)blockIdx.x * blockDim.x + threadIdx.x;
  float4 a = ((const float4*)z)[g];
  float4 b = ((const float4*)x0)[g];
  ((float4*)z)[g] = make_float4(2.f * a.x - b.x, 2.f * a.y - b.y,
                                2.f * a.z - b.z, 2.f * a.w - b.w);
}

// ---------------- GEMM: out (+)= X[131072,128] @ W_m[128,64] (+ bias) ---------
// W_m gathered from interleaved weight[640,64]: row f -> weight[(f*5+m_idx)]
// One 256-thread block = 8 waves; wave w computes rows [blk*128+w*16, +16) x 64 cols.
// f32 WMMA 16x16x4: A lane l holds row (l&15), K = khalf+{0,1}; B symmetric;
// C/D: VGPR v -> row (v + (l>>4)*8), col (l&15).
__global__ __launch_bounds__(256) void k_gemm_wmma(const float* __restrict__ X,
                                                   const float* __restrict__ W,
                                                   const float* __restrict__ bias,
                                                   float* __restrict__ out,
                                                   int m_idx, int accumulate) {
  __shared__ float lw[FDIM * OUTD];   // 32 KB: W_m slice, row-major [128][64]
  for (int i = threadIdx.x; i < FDIM * (OUTD / 4); i += 256) {
    int f = i >> 4;
    int o4 = i & 15;
    const float4* src = (const float4*)(W + (size_t)(f * 5 + m_idx) * OUTD);
    ((float4*)lw)[f * 16 + o4] = src[o4];
  }
  __syncthreads();

  int wave = threadIdx.x >> 5;
  int lane = threadIdx.x & 31;
  int nn    = lane & 15;          // col within 16-tile / A row
  int vhigh = lane >> 4;          // 0|1
  int khalf = vhigh * 2;          // K sub-offset {0,2}
  size_t row0 = (size_t)blockIdx.x * 128 + (size_t)wave * 16;

  v8f acc[4];
#pragma unroll
  for (int t = 0; t < 4; ++t) {
    int col = t * 16 + nn;
    if (accumulate) {
#pragma unroll
      for (int v = 0; v < 8; ++v)
        acc[t][v] = out[(row0 + (size_t)(v + vhigh * 8)) * OUTD + col];
    } else {
      float bv = bias[col];
#pragma unroll
      for (int v = 0; v < 8; ++v) acc[t][v] = bv;
    }
  }

  const float* xrow = X + (row0 + (size_t)nn) * FDIM;
#pragma unroll 4
  for (int k = 0; k < FDIM; k += 4) {
    v2f a;
    a.x = xrow[k + khalf];
    a.y = xrow[k + khalf + 1];
#pragma unroll
    for (int t = 0; t < 4; ++t) {
      v2f b;
      b.x = lw[(k + khalf) * OUTD + t * 16 + nn];
      b.y = lw[(k + khalf + 1) * OUTD + t * 16 + nn];
      acc[t] = __builtin_amdgcn_wmma_f32_16x16x4_f32(
          /*neg_a=*/false, a, /*neg_b=*/false, b,
          /*c_mod=*/(short)0, acc[t], /*reuse_a=*/false, /*reuse_b=*/false);
    }
  }

#pragma unroll
  for (int t = 0; t < 4; ++t) {
    int col = t * 16 + nn;
#pragma unroll
    for (int v = 0; v < 8; ++v)
      out[(row0 + (size_t)(v + vhigh * 8)) * OUTD + col] = acc[t][v];
  }
}

extern "C" void kernel_launch(void* const* d_in, const int* in_sizes, int n_in,
                              void* d_out, int out_size, void* d_ws, size_t ws_size,
                              hipStream_t stream) {
  const float* inp   = (const float*)d_in[0];
  const float* st    = (const float*)d_in[1];
  const int*   rows0 = (const int*)d_in[2];
  const int*   cols0 = (const int*)d_in[3];
  const float* vals0 = (const float*)d_in[4];
  const int*   rows1 = (const int*)d_in[5];
  const int*   cols1 = (const int*)d_in[6];
  const float* vals1 = (const float*)d_in[7];
  const float* wght  = (const float*)d_in[8];
  const float* bias  = (const float*)d_in[9];
  float* out = (float*)d_out;

  float* x0 = (float*)d_ws;
  float* y  = x0 + BUF_ELEMS;
  float* z  = y + BUF_ELEMS;

  int E0 = in_sizes[2];
  int E1 = in_sizes[5];

  dim3 blk(256);
  const int GRID_ELEM = (int)(BUF_ELEMS / 4 / 256);   // float4-vectorized elementwise
  const int GRID_GEMM = ROWS_TOTAL / 128;

  // x0 = concat(inputs, state); out = x0 @ W_0 + bias
  k_concat<<<GRID_ELEM, blk, 0, stream>>>(inp, st, x0);
  k_gemm_wmma<<<GRID_GEMM, blk, 0, stream>>>(x0, wght, bias, out, 0, 0);

  // support 0: x1 = A0 x0 ; x2 = 2 A0 x1 - x0
  k_zero<<<GRID_ELEM, blk, 0, stream>>>(y);
  k_spmm<<<E0, blk, 0, stream>>>(rows0, cols0, vals0, x0, y);
  k_gemm_wmma<<<GRID_GEMM, blk, 0, stream>>>(y, wght, bias, out, 1, 1);

  k_zero<<<GRID_ELEM, blk, 0, stream>>>(z);
  k_spmm<<<E0, blk, 0, stream>>>(rows0, cols0, vals0, y, z);
  k_cheb<<<GRID_ELEM, blk, 0, stream>>>(z, x0);
  k_gemm_wmma<<<GRID_GEMM, blk, 0, stream>>>(z, wght, bias, out, 2, 1);

  // support 1
  k_zero<<<GRID_ELEM, blk, 0, stream>>>(y);
  k_spmm<<<E1, blk, 0, stream>>>(rows1, cols1, vals1, x0, y);
  k_gemm_wmma<<<GRID_GEMM, blk, 0, stream>>>(y, wght, bias, out, 3, 1);

  k_zero<<<GRID_ELEM, blk, 0, stream>>>(z);
  k_spmm<<<E1, blk, 0, stream>>>(rows1, cols1, vals1, y, z);
  k_cheb<<<GRID_ELEM, blk, 0, stream>>>(z, x0);
  k_gemm_wmma<<<GRID_GEMM, blk, 0, stream>>>(z, wght, bias, out, 4, 1);
}